// MalwareGNN_71184787964257
// MI455X (gfx1250) — compile-verified
//
#include <hip/hip_runtime.h>
#include <hip/hip_bf16.h>
#include <cstddef>

// ---------------------------------------------------------------------------
// MalwareGNN on MI455X (gfx1250): 3x GCNConv+BN+ReLU, mean/max pool, MLP head.
// Dense GEMMs use V_WMMA_F32_16X16X4_F32 (fp32-exact matrix path).
// Edge aggregation uses HW fp32 global atomics; each node array is ~51MB so
// the whole scatter/gather working set is resident in the 192MB L2.
// All feature widths are compile-time constants -> no 64-bit div/mod per
// thread, fully unrolled WMMA k-loops, immediate strides.
// ---------------------------------------------------------------------------

#define NN 100000
#define NE 2000000
#define NB 128

typedef __attribute__((ext_vector_type(2))) float v2f;
typedef __attribute__((ext_vector_type(8))) float v8f;

__device__ __forceinline__ void atomic_add_f32(float* p, float v) {
  unsafeAtomicAdd(p, v);   // lowers to global_atomic_add_f32 (no-return HW atomic)
}

// ---------------- degree -----------------------------------------------------
__global__ void k_degree(const long long* __restrict__ dst, float* __restrict__ deg,
                         long long E) {
  long long i = blockIdx.x * (long long)blockDim.x + threadIdx.x;
  if (i < E) atomic_add_f32(&deg[dst[i]], 1.0f);
}

__global__ void k_deg_finalize(const float* __restrict__ deg, float* __restrict__ inv_sqrt,
                               float* __restrict__ inv_deg, int n) {
  int i = blockIdx.x * blockDim.x + threadIdx.x;
  if (i >= n) return;
  float d = deg[i] + 1.0f;             // +1 for self loop
  inv_sqrt[i] = rsqrtf(d);
  inv_deg[i]  = 1.0f / d;
}

// ---------------- WMMA fp32 GEMM: out[M,F] = A[M,K] @ W[K,F] (+bias, relu) ---
// One wave per 16x16 output tile; K,F compile-time -> fully unrolled WMMA chain.
// A frag (16x4): lane l<16 holds M=l K={k0,k0+1}; l>=16 holds M=l-16 K={k0+2,k0+3}.
// B frag (4x16): VGPR r, lane half h: K = k0 + r + 2h, N = l%16.
// C/D (16x16):  VGPR r: M = r + 8h, N = l%16.
template <int K, int F, int RELU>
__global__ void k_wmma_gemm(const float* __restrict__ A, const float* __restrict__ W,
                            const float* __restrict__ bias, float* __restrict__ out) {
  const int lane = threadIdx.x;          // 0..31 (wave32)
  const int half = lane >> 4;
  const int lm   = lane & 15;
  const long long row = (long long)blockIdx.x * 16 + lm;
  const int col = blockIdx.y * 16 + lm;
  const float* Arow = A + row * (long long)K;
  v8f acc = {};
#pragma unroll
  for (int k0 = 0; k0 < K; k0 += 4) {
    const int ka = k0 + 2 * half;
    v2f a, b;
    a.x = Arow[ka];
    a.y = Arow[ka + 1];
    b.x = W[(size_t)ka * F + col];
    b.y = W[(size_t)(ka + 1) * F + col];
    acc = __builtin_amdgcn_wmma_f32_16x16x4_f32(false, a, false, b,
                                                (short)0, acc, false, false);
  }
  const float bv = bias ? bias[col] : 0.0f;
  const long long obase = ((long long)blockIdx.x * 16 + 8 * half) * F + col;
#pragma unroll
  for (int r = 0; r < 8; ++r) {
    float v = acc[r] + bv;
    if (RELU) v = fmaxf(v, 0.0f);
    out[obase + (long long)r * F] = v;
  }
}

// ---------------- edge aggregation: agg[dst] += xw[src] * norm ---------------
// F/4 consecutive lanes per edge -> coalesced float4 gather + 4 HW atomics.
// F compile-time: e = idx >> log2(F/4), f = (idx & (F/4-1))*4  (no divides).
template <int F>
__global__ void k_edge_agg(const float* __restrict__ xw, const float* __restrict__ inv_sqrt,
                           const long long* __restrict__ src, const long long* __restrict__ dst,
                           float* __restrict__ agg, long long total) {
  long long idx = blockIdx.x * (long long)blockDim.x + threadIdx.x;
  if (idx >= total) return;
  constexpr int F4 = F >> 2;
  long long e = idx / F4;                 // F4 is a power-of-two constant -> shift
  int f = (int)(idx & (long long)(F4 - 1)) << 2;
  int s = (int)src[e];
  int d = (int)dst[e];
  float norm = inv_sqrt[s] * inv_sqrt[d];
  const float4 v = *(const float4*)(xw + (size_t)s * F + f);
  float* ap = agg + (size_t)d * F + f;
  atomic_add_f32(ap + 0, v.x * norm);
  atomic_add_f32(ap + 1, v.y * norm);
  atomic_add_f32(ap + 2, v.z * norm);
  atomic_add_f32(ap + 3, v.w * norm);
}

// ---------------- combine self-loop + bias, accumulate BN stats --------------
// blockDim.x == F; each block reduces ROWS rows locally, one atomic/feature.
template <int F>
__global__ void k_combine_stats(const float* __restrict__ xw, float* __restrict__ agg,
                                const float* __restrict__ bias, const float* __restrict__ inv_deg,
                                float* __restrict__ sum, float* __restrict__ sumsq,
                                int n, int rows) {
  int f  = threadIdx.x;
  int r0 = blockIdx.x * rows;
  int r1 = min(r0 + rows, n);
  float b = bias[f];
  float s = 0.0f, sq = 0.0f;
  for (int nd = r0; nd < r1; ++nd) {
    size_t i = (size_t)nd * F + f;
    float v = agg[i] + xw[i] * inv_deg[nd] + b;   // pre-BN activation
    agg[i] = v;                                   // in-place -> "pre" buffer
    s += v; sq += v * v;
  }
  atomic_add_f32(&sum[f], s);
  atomic_add_f32(&sumsq[f], sq);
}

// ---------------- batch-norm (biased var) + ReLU -----------------------------
template <int F>
__global__ void k_bn_relu(const float* __restrict__ pre, const float* __restrict__ sum,
                          const float* __restrict__ sumsq, const float* __restrict__ g,
                          const float* __restrict__ be, float* __restrict__ out,
                          int n) {
  long long idx = blockIdx.x * (long long)blockDim.x + threadIdx.x;
  if (idx >= (long long)n * F) return;
  int f = (int)(idx & (long long)(F - 1));        // F power of two
  float inv_n = 1.0f / (float)n;
  float mean  = sum[f] * inv_n;
  float var   = sumsq[f] * inv_n - mean * mean;
  float v = (pre[idx] - mean) * rsqrtf(var + 1e-5f) * g[f] + be[f];
  out[idx] = fmaxf(v, 0.0f);
}

// ---------------- pooling: per-graph sum / max / count (F = 64) --------------
// h >= 0 (post-ReLU), pmax zero-initialized -> int-bitcast atomicMax is exact.
__global__ void k_pool(const float* __restrict__ h, const long long* __restrict__ batch,
                       float* __restrict__ psum, float* __restrict__ pmax,
                       float* __restrict__ pcnt, int n) {
  constexpr int F = 64;
  long long idx = blockIdx.x * (long long)blockDim.x + threadIdx.x;
  if (idx >= (long long)n * F) return;
  int nd = (int)(idx >> 6);
  int f  = (int)(idx & 63);
  int b  = (int)batch[nd];
  float v = h[idx];
  atomic_add_f32(&psum[b * F + f], v);
  atomicMax((int*)&pmax[b * F + f], __float_as_int(v));
  if (f == 0) atomic_add_f32(&pcnt[b], 1.0f);
}

__global__ void k_emb(const float* __restrict__ psum, const float* __restrict__ pmax,
                      const float* __restrict__ pcnt, float* __restrict__ emb) {
  constexpr int F = 64;
  int idx = blockIdx.x * blockDim.x + threadIdx.x;      // NB * 2F
  if (idx >= NB * 2 * F) return;
  int b = idx >> 7;            // / (2F)
  int c = idx & 127;           // % (2F)
  emb[idx] = (c < F) ? psum[b * F + c] / pcnt[b] : pmax[b * F + (c - F)];
}

// ---------------- tiny fc3 (out features = 2) --------------------------------
__global__ void k_fc_small(const float* __restrict__ z, const float* __restrict__ W,
                           const float* __restrict__ bias, float* __restrict__ out) {
  constexpr int K = 32, O = 2;
  int idx = blockIdx.x * blockDim.x + threadIdx.x;      // NB * O
  if (idx >= NB * O) return;
  int b = idx >> 1, o = idx & 1;
  float s = bias[o];
#pragma unroll
  for (int k = 0; k < K; ++k) s += z[b * K + k] * W[k * O + o];
  out[idx] = s;
}

// ---------------------------------------------------------------------------
extern "C" void kernel_launch(void* const* d_in, const int* in_sizes, int n_in,
                              void* d_out, int out_size, void* d_ws, size_t ws_size,
                              hipStream_t stream) {
  (void)in_sizes; (void)n_in; (void)out_size; (void)ws_size;
  const float*     x    = (const float*)d_in[0];
  const long long* ei   = (const long long*)d_in[1];
  const long long* bat  = (const long long*)d_in[2];
  const float *W1 = (const float*)d_in[3],  *b1 = (const float*)d_in[4];
  const float *W2 = (const float*)d_in[5],  *b2 = (const float*)d_in[6];
  const float *W3 = (const float*)d_in[7],  *b3 = (const float*)d_in[8];
  const float *g1 = (const float*)d_in[9],  *be1 = (const float*)d_in[10];
  const float *g2 = (const float*)d_in[11], *be2 = (const float*)d_in[12];
  const float *g3 = (const float*)d_in[13], *be3 = (const float*)d_in[14];
  const float *fc1W = (const float*)d_in[15], *fc1b = (const float*)d_in[16];
  const float *fc2W = (const float*)d_in[17], *fc2b = (const float*)d_in[18];
  const float *fc3W = (const float*)d_in[19], *fc3b = (const float*)d_in[20];
  const long long* src = ei;
  const long long* dst = ei + NE;

  // ---- workspace layout (floats) ----
  float* ws = (float*)d_ws;
  float* deg      = ws;                       // N
  float* inv_sqrt = deg + NN;                 // N
  float* inv_deg  = inv_sqrt + NN;            // N
  float* xw       = inv_deg + NN;             // N*128
  float* agg      = xw  + (size_t)NN * 128;   // N*128  (becomes "pre" in place)
  float* hA       = agg + (size_t)NN * 128;   // N*128
  float* hB       = hA  + (size_t)NN * 128;   // N*128
  float* bnsum    = hB  + (size_t)NN * 128;   // 128
  float* bnsq     = bnsum + 128;              // 128
  float* psum     = bnsq  + 128;              // B*64
  float* pmax     = psum + NB * 64;           // B*64
  float* pcnt     = pmax + NB * 64;           // B
  float* z1       = pcnt + NB;                // B*64
  float* z2       = z1 + NB * 64;             // B*32

  float* logits = (float*)d_out;              // [B,2]
  float* emb    = logits + NB * 2;            // [B,128]

  const int TILE_M = NN / 16;                 // 6250, exact

  // ---- degrees ----
  hipMemsetAsync(deg, 0, NN * sizeof(float), stream);
  k_degree<<<(NE + 255) / 256, 256, 0, stream>>>(dst, deg, (long long)NE);
  k_deg_finalize<<<(NN + 255) / 256, 256, 0, stream>>>(deg, inv_sqrt, inv_deg, NN);

  // ---- layer 1: 12 -> 64 ----
  hipMemsetAsync(agg,   0, (size_t)NN * 64 * sizeof(float), stream);
  hipMemsetAsync(bnsum, 0, 256 * sizeof(float), stream);
  k_wmma_gemm<12, 64, 0><<<dim3(TILE_M, 4), 32, 0, stream>>>(x, W1, nullptr, xw);
  {
    long long tot = (long long)NE * 16;
    k_edge_agg<64><<<(unsigned)((tot + 255) / 256), 256, 0, stream>>>(
        xw, inv_sqrt, src, dst, agg, tot);
  }
  k_combine_stats<64><<<(NN + 63) / 64, 64, 0, stream>>>(xw, agg, b1, inv_deg,
                                                         bnsum, bnsq, NN, 64);
  k_bn_relu<64><<<(unsigned)(((long long)NN * 64 + 255) / 256), 256, 0, stream>>>(
      agg, bnsum, bnsq, g1, be1, hA, NN);

  // ---- layer 2: 64 -> 128 ----
  hipMemsetAsync(agg,   0, (size_t)NN * 128 * sizeof(float), stream);
  hipMemsetAsync(bnsum, 0, 256 * sizeof(float), stream);
  k_wmma_gemm<64, 128, 0><<<dim3(TILE_M, 8), 32, 0, stream>>>(hA, W2, nullptr, xw);
  {
    long long tot = (long long)NE * 32;
    k_edge_agg<128><<<(unsigned)((tot + 255) / 256), 256, 0, stream>>>(
        xw, inv_sqrt, src, dst, agg, tot);
  }
  k_combine_stats<128><<<(NN + 63) / 64, 128, 0, stream>>>(xw, agg, b2, inv_deg,
                                                           bnsum, bnsq, NN, 64);
  k_bn_relu<128><<<(unsigned)(((long long)NN * 128 + 255) / 256), 256, 0, stream>>>(
      agg, bnsum, bnsq, g2, be2, hB, NN);

  // ---- layer 3: 128 -> 64 ----
  hipMemsetAsync(agg,   0, (size_t)NN * 64 * sizeof(float), stream);
  hipMemsetAsync(bnsum, 0, 256 * sizeof(float), stream);
  k_wmma_gemm<128, 64, 0><<<dim3(TILE_M, 4), 32, 0, stream>>>(hB, W3, nullptr, xw);
  {
    long long tot = (long long)NE * 16;
    k_edge_agg<64><<<(unsigned)((tot + 255) / 256), 256, 0, stream>>>(
        xw, inv_sqrt, src, dst, agg, tot);
  }
  k_combine_stats<64><<<(NN + 63) / 64, 64, 0, stream>>>(xw, agg, b3, inv_deg,
                                                         bnsum, bnsq, NN, 64);
  k_bn_relu<64><<<(unsigned)(((long long)NN * 64 + 255) / 256), 256, 0, stream>>>(
      agg, bnsum, bnsq, g3, be3, hA, NN);

  // ---- pooling -> emb (written straight into d_out) ----
  hipMemsetAsync(psum, 0, (2 * NB * 64 + NB) * sizeof(float), stream);  // psum,pmax,pcnt
  k_pool<<<(unsigned)(((long long)NN * 64 + 255) / 256), 256, 0, stream>>>(
      hA, bat, psum, pmax, pcnt, NN);
  k_emb<<<(NB * 128 + 255) / 256, 256, 0, stream>>>(psum, pmax, pcnt, emb);

  // ---- MLP head: fc1/fc2 via WMMA with fused bias+ReLU, fc3 scalar ----
  k_wmma_gemm<128, 64, 1><<<dim3(NB / 16, 4), 32, 0, stream>>>(emb, fc1W, fc1b, z1);
  k_wmma_gemm<64, 32, 1><<<dim3(NB / 16, 2), 32, 0, stream>>>(z1,  fc2W, fc2b, z2);
  k_fc_small<<<1, 256, 0, stream>>>(z2, fc3W, fc3b, logits);
}